// AffineTransform_81492709474313
// MI455X (gfx1250) — compile-verified
//
#include <hip/hip_runtime.h>
#include <math.h>

typedef __attribute__((ext_vector_type(2))) float v2f;
typedef __attribute__((ext_vector_type(8))) float v8f;

static constexpr int kB = 32, kC = 4, kH = 512, kW = 512;
static constexpr float kPI = 3.14159265358979323846f;

// d_out layout (float element offsets), reference return order:
static constexpr size_t OFF_T     = 0;                                    // transformed (32,4,512,512)
static constexpr size_t OFF_MAT   = (size_t)kB * kC * kH * kW;            // 33554432
static constexpr size_t OFF_INV   = OFF_MAT + (size_t)kB * 9;             // +288
static constexpr size_t OFF_GRID  = OFF_INV + (size_t)kB * 9;             // +288
static constexpr size_t OFF_IGRID = OFF_GRID + (size_t)kB * kH * kW * 2;  // +16777216

__device__ __forceinline__ int iclamp(int v, int lo, int hi) {
    return v < lo ? lo : (v > hi ? hi : v);
}

// ---------------- Kernel 1: build 3x3 mats + inverses (32 batches) ----------------
__global__ void build_mats_kernel(const float* __restrict__ fc2,
                                  float* __restrict__ out,
                                  float* __restrict__ ws12) {
    int b = threadIdx.x;
    if (b >= kB) return;
    const float* f = fc2 + b * 7;

    float theta = fminf(fmaxf(f[0] * 0.3f, -1.0f), 1.0f) * kPI;
    float sx    = fminf(fmaxf(f[1] * 0.3f + 1.0f, 0.0f), 5.0f);
    float sy    = fminf(fmaxf(f[2] * 0.3f + 1.0f, 0.0f), 5.0f);
    float tx    = f[3] * 0.3f;
    float ty    = f[4] * 0.3f;
    float shxy  = fminf(fmaxf(f[5] * 0.3f, -1.0f), 1.0f) * kPI;
    float shyx  = fminf(fmaxf(f[6] * 0.3f, -1.0f), 1.0f) * kPI;

    float c = cosf(theta), s = sinf(theta);
    // M = T @ Sh @ Sc @ R  (closed form; bottom row [0,0,1])
    float m00 = sx * c + shxy * sy * s;
    float m01 = -sx * s + shxy * sy * c;
    float m02 = tx;
    float m10 = shyx * sx * c + sy * s;
    float m11 = -shyx * sx * s + sy * c;
    float m12 = ty;

    float det = m00 * m11 - m01 * m10;
    float id  = 1.0f / det;
    float i00 =  m11 * id;
    float i01 = -m01 * id;
    float i02 = (m01 * m12 - m11 * m02) * id;
    float i10 = -m10 * id;
    float i11 =  m00 * id;
    float i12 = (m10 * m02 - m00 * m12) * id;

    float* m = out + OFF_MAT + (size_t)b * 9;
    m[0] = m00; m[1] = m01; m[2] = m02;
    m[3] = m10; m[4] = m11; m[5] = m12;
    m[6] = 0.0f; m[7] = 0.0f; m[8] = 1.0f;

    float* im = out + OFF_INV + (size_t)b * 9;
    im[0] = i00; im[1] = i01; im[2] = i02;
    im[3] = i10; im[4] = i11; im[5] = i12;
    im[6] = 0.0f; im[7] = 0.0f; im[8] = 1.0f;

    float* w = ws12 + (size_t)b * 12;
    w[0] = m00; w[1] = m01; w[2]  = m02;
    w[3] = m10; w[4] = m11; w[5]  = m12;
    w[6] = i00; w[7] = i01; w[8]  = i02;
    w[9] = i10; w[10] = i11; w[11] = i12;
}

// ---------------- Kernel 2: WMMA grid generation + bilinear sample ----------------
// One thread per pixel; each wave32 owns 32 contiguous pixels of one row of one batch.
// Grid coords for the wave's 32 pixels are produced by two V_WMMA_F32_16X16X4_F32:
//   A (16x4): rows 0..3 = [mat r0; mat r1; inv r0; inv r1], rest 0   (wave-uniform)
//   B (4x16): col n = [x_n, y_n, 1, 0]                               (16 pixels/WMMA)
//   D (16x16): row0=gx row1=gy row2=igx row3=igy per pixel column.
__global__ __launch_bounds__(256) void warp_sample_kernel(const float* __restrict__ src,
                                                          const float* __restrict__ ws12,
                                                          float* __restrict__ out) {
    const int tid  = blockIdx.x * 256 + threadIdx.x;   // 0 .. 32*512*512-1
    const int lane = threadIdx.x & 31;
    const int b    = tid >> 18;                        // / (512*512)
    const int rem  = tid & 0x3FFFF;
    const int h    = rem >> 9;
    const int w    = rem & 511;

    const float xn = (2.0f * (float)w + 1.0f) * (1.0f / 512.0f) - 1.0f;
    const float yn = (2.0f * (float)h + 1.0f) * (1.0f / 512.0f) - 1.0f;

    // --- A matrix (f32 16x4 layout: lane<16 -> {K0,K1}, lane>=16 -> {K2,K3}, M = lane%16) ---
    const float* mb = ws12 + (size_t)b * 12;
    const int  r   = lane & 15;
    const bool rOK = (r < 4);
    const bool lo  = (lane < 16);
    const int  i0  = rOK ? (r * 3 + (lo ? 0 : 2)) : 0;
    const int  i1  = (rOK && lo) ? (r * 3 + 1) : 0;
    v2f A;
    A[0] = rOK ? mb[i0] : 0.0f;
    A[1] = (rOK && lo) ? mb[i1] : 0.0f;

    // --- B tiles (f32 4x16 layout: N = lane%16; lane<16 -> {K0,K1}, lane>=16 -> {K2,K3}) ---
    v2f B1;                                      // pixels 0..15 of the wave
    B1[0] = lo ? xn : 1.0f;
    B1[1] = lo ? yn : 0.0f;
    const float xh = __shfl(xn, (lane & 15) + 16, 32);
    const float yh = __shfl(yn, (lane & 15) + 16, 32);
    v2f B2;                                      // pixels 16..31 of the wave
    B2[0] = lo ? xh : 1.0f;
    B2[1] = lo ? yh : 0.0f;

    v8f Cz = {0.f, 0.f, 0.f, 0.f, 0.f, 0.f, 0.f, 0.f};
    v8f D1 = __builtin_amdgcn_wmma_f32_16x16x4_f32(false, A, false, B1, (short)0, Cz, false, false);
    v8f D2 = __builtin_amdgcn_wmma_f32_16x16x4_f32(false, A, false, B2, (short)0, Cz, false, false);

    // D row j lives in VGPR j, lanes 0..15 (col = lane). Lanes 16..31 pull their pixel
    // (tile-2 column lane-16) from source lane lane&15.
    const float gx2  = __shfl(D2[0], lane & 15, 32);
    const float gy2  = __shfl(D2[1], lane & 15, 32);
    const float igx2 = __shfl(D2[2], lane & 15, 32);
    const float igy2 = __shfl(D2[3], lane & 15, 32);
    const float gx  = lo ? D1[0] : gx2;
    const float gy  = lo ? D1[1] : gy2;
    const float igx = lo ? D1[2] : igx2;
    const float igy = lo ? D1[3] : igy2;

    // --- grid / inv_grid stores (coalesced float2) ---
    ((float2*)(out + OFF_GRID))[tid]  = make_float2(gx, gy);
    ((float2*)(out + OFF_IGRID))[tid] = make_float2(igx, igy);

    // --- bilinear sample of 4 channels through (gx, gy) ---
    const float ix  = ((gx + 1.0f) * 512.0f - 1.0f) * 0.5f;
    const float iy  = ((gy + 1.0f) * 512.0f - 1.0f) * 0.5f;
    const float x0f = floorf(ix);
    const float y0f = floorf(iy);
    const float wx  = ix - x0f;
    const float wy  = iy - y0f;
    const int x0 = (int)x0f, y0 = (int)y0f;
    const int x1 = x0 + 1,  y1 = y0 + 1;

    const float v00m = ((x0 >= 0) & (x0 < 512) & (y0 >= 0) & (y0 < 512)) ? 1.0f : 0.0f;
    const float v01m = ((x1 >= 0) & (x1 < 512) & (y0 >= 0) & (y0 < 512)) ? 1.0f : 0.0f;
    const float v10m = ((x0 >= 0) & (x0 < 512) & (y1 >= 0) & (y1 < 512)) ? 1.0f : 0.0f;
    const float v11m = ((x1 >= 0) & (x1 < 512) & (y1 >= 0) & (y1 < 512)) ? 1.0f : 0.0f;
    const int x0c = iclamp(x0, 0, 511), x1c = iclamp(x1, 0, 511);
    const int y0c = iclamp(y0, 0, 511), y1c = iclamp(y1, 0, 511);

    const float w00 = (1.0f - wx) * (1.0f - wy);
    const float w01 = wx * (1.0f - wy);
    const float w10 = (1.0f - wx) * wy;
    const float w11 = wx * wy;

    const size_t plane = (size_t)kH * kW;
    const float* sb = src + (size_t)b * kC * plane;
    const int o00 = y0c * 512 + x0c;
    const int o01 = y0c * 512 + x1c;
    const int o10 = y1c * 512 + x0c;
    const int o11 = y1c * 512 + x1c;

#pragma unroll
    for (int c = 0; c < kC; ++c) {
        const float* img = sb + (size_t)c * plane;
        const float v00 = img[o00] * v00m;
        const float v01 = img[o01] * v01m;
        const float v10 = img[o10] * v10m;
        const float v11 = img[o11] * v11m;
        out[OFF_T + ((size_t)(b * kC + c)) * plane + rem] =
            v00 * w00 + v01 * w01 + v10 * w10 + v11 * w11;
    }
}

extern "C" void kernel_launch(void* const* d_in, const int* in_sizes, int n_in,
                              void* d_out, int out_size, void* d_ws, size_t ws_size,
                              hipStream_t stream) {
    const float* src = (const float*)d_in[0];   // (32,4,512,512) f32
    const float* fc2 = (const float*)d_in[1];   // (32,7) f32
    float* out = (float*)d_out;
    float* ws  = (float*)d_ws;                  // needs 32*12 floats = 1536 B

    build_mats_kernel<<<1, 32, 0, stream>>>(fc2, out, ws);

    const int nPix   = kB * kH * kW;            // 8,388,608
    const int blocks = nPix / 256;              // 32,768
    warp_sample_kernel<<<blocks, 256, 0, stream>>>(src, ws, out);
}